// Ternary_2138893714279
// MI455X (gfx1250) — compile-verified
//
#include <hip/hip_runtime.h>
#include <stdint.h>

#ifndef __has_builtin
#define __has_builtin(x) 0
#endif

#define IMG_N   8
#define IMG_H   512
#define IMG_W   512
#define PAD     3
#define TILE    32
#define GDIM    (TILE + 2 * PAD)      // 38
#define GPLANE  (GDIM * GDIM)         // 1444 (compact rgb staging plane)
#define GSTRIDE 40                    // padded row stride for gray tiles
#define GTOT    (GDIM * GSTRIDE)      // 1520

typedef __attribute__((address_space(1))) int g_int_t;
typedef __attribute__((address_space(3))) int l_int_t;

// ---- CDNA5 async global->LDS copy (ASYNCcnt path) -------------------------
__device__ __forceinline__ void async_copy_f32(const float* gsrc, float* ldst) {
#if __has_builtin(__builtin_amdgcn_global_load_async_to_lds_b32)
  __builtin_amdgcn_global_load_async_to_lds_b32(
      (g_int_t*)(uintptr_t)gsrc,
      (l_int_t*)(uint32_t)(uintptr_t)ldst,   // generic LDS addr low 32b == LDS offset
      /*offset=*/0, /*cpol=*/0);
#else
  uint32_t loff = (uint32_t)(uintptr_t)ldst;
  uint64_t ga   = (uint64_t)(uintptr_t)gsrc;
  asm volatile("global_load_async_to_lds_b32 %0, %1, off"
               :: "v"(loff), "v"(ga) : "memory");
#endif
}

__device__ __forceinline__ void async_wait0() {
#if __has_builtin(__builtin_amdgcn_s_wait_asynccnt)
  __builtin_amdgcn_s_wait_asynccnt(0);
#else
  asm volatile("s_wait_asynccnt 0" ::: "memory");
#endif
}

// ---------------------------------------------------------------------------
__global__ __launch_bounds__(256)
void census_hamming_kernel(const float* __restrict__ img0,
                           const float* __restrict__ img1,
                           float* __restrict__ out) {
  __shared__ float s_rgb[6 * GPLANE];   // [img*3+c][38*38]   ~34.7 KB
  __shared__ float s_gray[2 * GTOT];    // [img][38*40]       ~12.2 KB

  const int tid = threadIdx.x;           // 0..255  (8 wave32 waves)
  const int ox  = blockIdx.x * TILE;
  const int oy  = blockIdx.y * TILE;
  const int n   = blockIdx.z;

  // ---- Stage 1: async-copy rgb halo tiles (2 imgs x 3 ch x 38x38) to LDS --
#pragma unroll
  for (int pl = 0; pl < 6; ++pl) {
    const float* base = (pl < 3) ? img0 : img1;
    const int c = (pl < 3) ? pl : pl - 3;
    const size_t plane_off = ((size_t)(n * 3 + c)) * IMG_H * IMG_W;
    for (int pos = tid; pos < GPLANE; pos += 256) {
      const int ly = pos / GDIM;
      const int lx = pos - ly * GDIM;
      int gx = ox + lx - PAD;
      int gy = oy + ly - PAD;
      gx = min(max(gx, 0), IMG_W - 1);    // clamp; zero-fill handled later
      gy = min(max(gy, 0), IMG_H - 1);
      async_copy_f32(base + plane_off + (size_t)gy * IMG_W + gx,
                     &s_rgb[pl * GPLANE + pos]);
    }
  }
  async_wait0();
  __syncthreads();

  // ---- Stage 2: rgb -> gray in LDS (zero outside the image = zero padding)
  for (int e = tid; e < 2 * GPLANE; e += 256) {
    const int img = (e >= GPLANE) ? 1 : 0;
    const int pos = e - img * GPLANE;
    const int ly = pos / GDIM;
    const int lx = pos - ly * GDIM;
    const int gx = ox + lx - PAD;
    const int gy = oy + ly - PAD;
    const bool valid = (gx >= 0) & (gx < IMG_W) & (gy >= 0) & (gy < IMG_H);
    const float r = s_rgb[(img * 3 + 0) * GPLANE + pos];
    const float g = s_rgb[(img * 3 + 1) * GPLANE + pos];
    const float b = s_rgb[(img * 3 + 2) * GPLANE + pos];
    const float gray = fmaf(0.2989f, r, fmaf(0.587f, g, 0.114f * b));
    s_gray[img * GTOT + ly * GSTRIDE + lx] = valid ? gray : 0.0f;
  }
  __syncthreads();

  // ---- Stage 3: 49-tap census transform + robust hamming mean -------------
  const int tx = tid & 31;          // 32 x 8 thread tile
  const int ty = tid >> 5;
  const int x  = ox + tx;
  const int lc = tx + PAD;

#pragma unroll 1
  for (int k = 0; k < 4; ++k) {
    const int y  = oy + ty + k * 8;
    const int lr = ty + k * 8 + PAD;
    const int cidx = lr * GSTRIDE + lc;
    const float g0c = s_gray[cidx];
    const float g1c = s_gray[GTOT + cidx];

    float acc = 0.0f;
#pragma unroll
    for (int dy = -PAD; dy <= PAD; ++dy) {
#pragma unroll
      for (int dx = -PAD; dx <= PAD; ++dx) {
        const int idx = cidx + dy * GSTRIDE + dx;
        const float d0 = s_gray[idx] - g0c;
        const float d1 = s_gray[GTOT + idx] - g1c;
        // t = d * rsqrt(0.81 + d*d)   (v_rsq_f32, co-executes on TRANS unit)
        const float t0 = d0 * __builtin_amdgcn_rsqf(fmaf(d0, d0, 0.81f));
        const float t1 = d1 * __builtin_amdgcn_rsqf(fmaf(d1, d1, 0.81f));
        const float df = t0 - t1;
        const float s  = df * df;
        acc = fmaf(s, __builtin_amdgcn_rcpf(s + 0.1f), acc);
      }
    }
    const float mask =
        (x >= 1 && x < IMG_W - 1 && y >= 1 && y < IMG_H - 1) ? (1.0f / 49.0f)
                                                             : 0.0f;
    out[((size_t)n * IMG_H + y) * IMG_W + x] = acc * mask;
  }
}

// ---------------------------------------------------------------------------
extern "C" void kernel_launch(void* const* d_in, const int* in_sizes, int n_in,
                              void* d_out, int out_size, void* d_ws,
                              size_t ws_size, hipStream_t stream) {
  (void)in_sizes; (void)n_in; (void)out_size; (void)d_ws; (void)ws_size;
  const float* img0 = (const float*)d_in[0];
  const float* img1 = (const float*)d_in[1];
  float* out = (float*)d_out;

  dim3 grid(IMG_W / TILE, IMG_H / TILE, IMG_N);   // 16 x 16 x 8 = 2048 blocks
  dim3 block(256, 1, 1);                          // 8 wave32 waves
  census_hamming_kernel<<<grid, block, 0, stream>>>(img0, img1, out);
}